// TimeSeriesPredictor_87900800680428
// MI455X (gfx1250) — compile-verified
//
#include <hip/hip_runtime.h>

// ---- vector types -----------------------------------------------------------
typedef float v2f __attribute__((ext_vector_type(2)));
typedef float v8f __attribute__((ext_vector_type(8)));
typedef float f4  __attribute__((ext_vector_type(4)));
typedef f4 uf4 __attribute__((aligned(4)));   // 4-byte-aligned 128-bit load

// fast sigmoid / tanh on the serial critical path (v_exp2 + v_rcp)
__device__ __forceinline__ float fsig(float x) {
    return __builtin_amdgcn_rcpf(1.0f + __builtin_amdgcn_exp2f(-1.442695041f * x));
}
__device__ __forceinline__ float ftanh(float x) {
    // tanh(x) = 2*sigmoid(2x) - 1
    return 2.0f * fsig(2.0f * x) - 1.0f;
}

// One wave32 handles 16 batch elements.
//   D/C tile (16x16 f32): VGPR r, lane n(0..15) -> gate r of batch n  (gates on M, batch on N)
//   A1 (16x4): rows 0..7 = W_ih[:,0:4]            (rows 8..15 zero)
//   A2 (16x4): rows 0..7 = [W_hh[:,0] W_hh[:,1] W_ih[:,4] 0]
//   B1 (4x16): x[t][0..3]^T for the 16 batches;  B2 (4x16): [h0; h1; x[t][4]; 0]
// B layout (wave32): VGPR0 lanes0-15 = K0, lanes16-31 = K2; VGPR1 lanes0-15 = K1, lanes16-31 = K3.
__global__ __launch_bounds__(128)
void TimeSeriesPredictor_lstm_wmma(const float* __restrict__ xs,
                                   const float* __restrict__ W_ih,
                                   const float* __restrict__ W_hh,
                                   const float* __restrict__ b_ih,
                                   const float* __restrict__ b_hh,
                                   const float* __restrict__ W_fc,
                                   const float* __restrict__ b_fc,
                                   float* __restrict__ out,
                                   int B, int T)
{
    const int lane = threadIdx.x & 31;
    const int wave = (int)((blockIdx.x * blockDim.x + threadIdx.x) >> 5);
    const int base = wave * 16;
    if (base >= B) return;                 // wave-uniform: EXEC stays full for live waves

    const int  m  = lane & 15;             // M row (A) / N column (B,D)
    const bool hi = lane >= 16;            // upper half-wave carries K=2,3 slices

    // ---- constant A operands (loaded once) ----------------------------------
    float a1_0 = 0.f, a1_1 = 0.f, a2_0 = 0.f, a2_1 = 0.f;
    if (m < 8) {                           // gate rows 0..7; rows 8..15 stay zero
        a1_0 = W_ih[m * 5 + (hi ? 2 : 0)];
        a1_1 = W_ih[m * 5 + (hi ? 3 : 1)];
        a2_0 = hi ? W_ih[m * 5 + 4] : W_hh[m * 2 + 0];
        a2_1 = hi ? 0.0f             : W_hh[m * 2 + 1];
    }
    const v2f A1 = { a1_0, a1_1 };
    const v2f A2 = { a2_0, a2_1 };

    // combined bias broadcast into the C accumulator (rows = gates)
    v8f Cb;
#pragma unroll
    for (int r = 0; r < 8; ++r) Cb[r] = b_ih[r] + b_hh[r];

    const float wfc0 = W_fc[0], wfc1 = W_fc[1], bfc = b_fc[0];

    const int    b    = base + m;
    const int    bsafe = (b < B) ? b : (B - 1);
    const float* xp   = xs + (size_t)bsafe * (size_t)T * 5;
    float*       yp   = out + (size_t)bsafe * (size_t)T;
    const bool   doStore = (!hi) && (b < B);

    float h0 = 0.f, h1 = 0.f, c0 = 0.f, c1 = 0.f;

    for (int t = 0; t < T; ++t) {
        // per-lane stream of this batch's x row (20 B/step, contiguous over the loop)
        f4    xv = *(const uf4*)xp;
        float x4 = xp[4];
        xp += 5;
        __builtin_prefetch(xp + 5 * 16, 0, 0);   // global_prefetch_b8, ~16 steps ahead

        // B operands for this step
        const v2f B1 = { hi ? xv.z : xv.x, hi ? xv.w : xv.y };   // x[0..3]^T
        const v2f B2 = { hi ? x4   : h0,   hi ? 0.0f : h1   };   // [h0;h1;x4;0]

        // gates[8 x 16] = W_ih[:,0:4]*x + bias, then += W_hh*h + W_ih[:,4]*x4
        v8f acc = __builtin_amdgcn_wmma_f32_16x16x4_f32(
                      false, A1, false, B1, (short)0, Cb, false, false);
        v8f g   = __builtin_amdgcn_wmma_f32_16x16x4_f32(
                      false, A2, false, B2, (short)0, acc, false, false);

        // per-lane LSTM cell update (PyTorch gate order i,f,g,o; H=2)
        const float i0 = fsig(g[0]),  i1 = fsig(g[1]);
        const float f0 = fsig(g[2]),  f1 = fsig(g[3]);
        const float z0 = ftanh(g[4]), z1 = ftanh(g[5]);
        const float o0 = fsig(g[6]),  o1 = fsig(g[7]);
        c0 = f0 * c0 + i0 * z0;
        c1 = f1 * c1 + i1 * z1;
        h0 = o0 * ftanh(c0);
        h1 = o1 * ftanh(c1);

        if (doStore) yp[t] = h0 * wfc0 + h1 * wfc1 + bfc;   // final fc layer, fused
    }
}

extern "C" void kernel_launch(void* const* d_in, const int* in_sizes, int n_in,
                              void* d_out, int out_size, void* d_ws, size_t ws_size,
                              hipStream_t stream) {
    (void)n_in; (void)d_ws; (void)ws_size; (void)out_size;
    const int T = 2048, I = 5;
    const int B = in_sizes[0] / (T * I);   // 4096

    const float* xs   = (const float*)d_in[0];
    const float* W_ih = (const float*)d_in[1];
    const float* W_hh = (const float*)d_in[2];
    const float* b_ih = (const float*)d_in[3];
    const float* b_hh = (const float*)d_in[4];
    const float* W_fc = (const float*)d_in[5];
    const float* b_fc = (const float*)d_in[6];
    float*       out  = (float*)d_out;

    const int waves   = (B + 15) / 16;      // 256 waves
    const int threads = waves * 32;         // 8192 threads
    const int block   = 128;                // 4 waves / workgroup
    const int grid    = (threads + block - 1) / block;

    TimeSeriesPredictor_lstm_wmma<<<grid, block, 0, stream>>>(
        xs, W_ih, W_hh, b_ih, b_hh, W_fc, b_fc, out, B, T);
}